// RegressionBranch_50337016709815
// MI455X (gfx1250) — compile-verified
//
#include <hip/hip_runtime.h>
#include <hip/hip_bf16.h>
#include <math.h>

typedef __attribute__((ext_vector_type(2))) float v2f;
typedef __attribute__((ext_vector_type(8))) float v8f;

#define FOUT 32           // all hidden/out dims are 32
#define NEG_SLOPE 0.2f

// ---------------------------------------------------------------------------
// sign-aware float atomic max (works for negative values, init with -inf)
// ---------------------------------------------------------------------------
__device__ __forceinline__ void atomicMaxF(float* addr, float val) {
    if (val >= 0.0f) {
        atomicMax((int*)addr, __float_as_int(val));
    } else {
        atomicMin((unsigned int*)addr, __float_as_uint(val));
    }
}

// ---------------------------------------------------------------------------
// GEMM  P[N x 32] = X[N x K] @ W[K x 32]  (+bias)  via V_WMMA_F32_16X16X4_F32
// One wave per 16-row tile; two 16-col WMMA accumulators cover the 32 cols.
// N is a multiple of 16 (100000 = 6250*16). All branches wave-uniform.
// ---------------------------------------------------------------------------
__global__ void gat_gemm_wmma_f32(const float* __restrict__ X,
                                  const float* __restrict__ W,
                                  const float* __restrict__ bias,
                                  float* __restrict__ P,
                                  int N, int K) {
    const int wavesPerBlock = blockDim.x >> 5;
    const int waveId  = threadIdx.x >> 5;
    const int lane    = threadIdx.x & 31;
    const int half    = lane >> 4;      // 0: lanes 0-15, 1: lanes 16-31
    const int l16     = lane & 15;
    const int rowTile = blockIdx.x * wavesPerBlock + waveId;
    const int numTiles = N >> 4;
    if (rowTile >= numTiles) return;    // wave-uniform guard
    const int rowBase = rowTile << 4;

    v8f c0 = {};  // cols 0..15
    v8f c1 = {};  // cols 16..31

    for (int k0 = 0; k0 < K; k0 += 4) {
        // A fragment (16x4 f32): lanes 0-15 hold K=k0..k0+1, lanes 16-31 K=k0+2..k0+3
        const float* arow = X + (size_t)(rowBase + l16) * K + k0 + half * 2;
        v2f a;
        a.x = arow[0];
        a.y = arow[1];
        // B fragments (4x16 f32): mirror layout, N in place of M
        const float* b0 = W + (size_t)(k0 + half * 2) * FOUT;
        v2f bA, bB;
        bA.x = b0[l16];             bA.y = b0[FOUT + l16];
        bB.x = b0[16 + l16];        bB.y = b0[FOUT + 16 + l16];

        c0 = __builtin_amdgcn_wmma_f32_16x16x4_f32(false, a, false, bA,
                                                   (short)0, c0, false, false);
        c1 = __builtin_amdgcn_wmma_f32_16x16x4_f32(false, a, false, bB,
                                                   (short)0, c1, false, false);
    }

    // C/D layout: VGPR r: lanes 0-15 -> (M=r, N=lane); lanes 16-31 -> (M=r+8, N=lane-16)
    #pragma unroll
    for (int r = 0; r < 8; ++r) {
        int row = rowBase + r + half * 8;
        float v0 = c0[r];
        float v1 = c1[r];
        if (bias) { v0 += bias[l16]; v1 += bias[16 + l16]; }
        P[(size_t)row * FOUT + l16]      = v0;
        P[(size_t)row * FOUT + 16 + l16] = v1;
    }
}

// ---------------------------------------------------------------------------
// Per-node prep: a_src = h·att_src, a_dst = h·att_dst ; init m/-inf, denom/0, acc/0
// ---------------------------------------------------------------------------
__global__ void gat_node_prep(const float* __restrict__ h,
                              const float* __restrict__ avs,
                              const float* __restrict__ avd,
                              float* __restrict__ asrc,
                              float* __restrict__ adst,
                              float* __restrict__ m,
                              float* __restrict__ denom,
                              float* __restrict__ acc,
                              int N) {
    int n = blockIdx.x * blockDim.x + threadIdx.x;
    if (n >= N) return;
    const float* hr = h + (size_t)n * FOUT;
    float* ar = acc + (size_t)n * FOUT;
    float ss = 0.0f, sd = 0.0f;
    #pragma unroll
    for (int f = 0; f < FOUT; ++f) {
        float v = hr[f];
        ss += v * avs[f];
        sd += v * avd[f];
        ar[f] = 0.0f;
    }
    asrc[n]  = ss;
    adst[n]  = sd;
    m[n]     = -__builtin_inff();
    denom[n] = 0.0f;
}

// ---------------------------------------------------------------------------
// Edge pass 1: e = LeakyReLU(a_src[src]+a_dst[dst]); store e; segment-max into m
// Edges e >= E are implicit self loops (src = dst = e - E).
// ---------------------------------------------------------------------------
__global__ void gat_edge_att(const long long* __restrict__ ei, int E, int Etot,
                             const float* __restrict__ asrc,
                             const float* __restrict__ adst,
                             float* __restrict__ att,
                             float* __restrict__ m) {
    int e = blockIdx.x * blockDim.x + threadIdx.x;
    if (e >= Etot) return;
    int s, d;
    if (e < E) { s = (int)ei[e]; d = (int)ei[(size_t)E + e]; }
    else       { s = d = e - E; }
    float v = asrc[s] + adst[d];
    v = (v > 0.0f) ? v : NEG_SLOPE * v;
    att[e] = v;
    atomicMaxF(&m[d], v);
}

// ---------------------------------------------------------------------------
// Edge pass 2: ex = exp(e - m[dst]); att <- ex; segment-sum into denom
// ---------------------------------------------------------------------------
__global__ void gat_edge_exp(const long long* __restrict__ ei, int E, int Etot,
                             const float* __restrict__ m,
                             float* __restrict__ att,
                             float* __restrict__ denom) {
    int e = blockIdx.x * blockDim.x + threadIdx.x;
    if (e >= Etot) return;
    int d = (e < E) ? (int)ei[(size_t)E + e] : (e - E);
    float ex = __expf(att[e] - m[d]);
    att[e] = ex;
    atomicAdd(&denom[d], ex);
}

// ---------------------------------------------------------------------------
// Edge pass 3: acc[dst] += h[src] * (ex / denom[dst]).  One wave32 per edge,
// lane == feature channel (FOUT == 32 == warpSize on gfx1250).
// ---------------------------------------------------------------------------
__global__ void gat_edge_scatter(const long long* __restrict__ ei, int E, int Etot,
                                 const float* __restrict__ att,
                                 const float* __restrict__ denom,
                                 const float* __restrict__ h,
                                 float* __restrict__ acc) {
    int e = blockIdx.x * (blockDim.x >> 5) + (threadIdx.x >> 5);
    int f = threadIdx.x & 31;
    if (e >= Etot) return;          // wave-uniform
    int s, d;
    if (e < E) { s = (int)ei[e]; d = (int)ei[(size_t)E + e]; }
    else       { s = d = e - E; }
    float alpha = att[e] / (denom[d] + 1e-16f);
    atomicAdd(&acc[(size_t)d * FOUT + f], h[(size_t)s * FOUT + f] * alpha);
}

// ---------------------------------------------------------------------------
// Finish a layer: hout = relu(acc + bias)
// ---------------------------------------------------------------------------
__global__ void gat_finish(const float* __restrict__ acc,
                           const float* __restrict__ bias,
                           float* __restrict__ hout,
                           int total) {
    int i = blockIdx.x * blockDim.x + threadIdx.x;
    if (i >= total) return;
    hout[i] = fmaxf(acc[i] + bias[i & (FOUT - 1)], 0.0f);
}

// ---------------------------------------------------------------------------
// Host-side orchestration
// ---------------------------------------------------------------------------
static inline size_t alignUp(size_t x) { return (x + 255) & ~(size_t)255; }

extern "C" void kernel_launch(void* const* d_in, const int* in_sizes, int n_in,
                              void* d_out, int out_size, void* d_ws, size_t ws_size,
                              hipStream_t stream) {
    const float*     x   = (const float*)d_in[0];      // [N, 128]
    const long long* ei  = (const long long*)d_in[1];  // [2, E] int64
    const float*     W1  = (const float*)d_in[2];      // [128, 32]
    const float*     as1 = (const float*)d_in[3];
    const float*     ad1 = (const float*)d_in[4];
    const float*     b1  = (const float*)d_in[5];
    const float*     W2  = (const float*)d_in[6];      // [32, 32]
    const float*     as2 = (const float*)d_in[7];
    const float*     ad2 = (const float*)d_in[8];
    const float*     b2  = (const float*)d_in[9];
    const float*     Wr  = (const float*)d_in[10];     // [32, 32]
    const float*     br  = (const float*)d_in[11];
    float* out = (float*)d_out;

    const int Fin  = 128;
    const int N    = in_sizes[0] / Fin;        // 100000
    const int E    = in_sizes[1] / 2;          // 3200000
    const int Etot = E + N;                    // + self loops

    // workspace carve-out
    char* w = (char*)d_ws;
    float* P     = (float*)w; w += alignUp((size_t)N * FOUT * sizeof(float));
    float* acc   = (float*)w; w += alignUp((size_t)N * FOUT * sizeof(float));
    float* Hbuf  = (float*)w; w += alignUp((size_t)N * FOUT * sizeof(float));
    float* asrc  = (float*)w; w += alignUp((size_t)N * sizeof(float));
    float* adst  = (float*)w; w += alignUp((size_t)N * sizeof(float));
    float* mbuf  = (float*)w; w += alignUp((size_t)N * sizeof(float));
    float* dens  = (float*)w; w += alignUp((size_t)N * sizeof(float));
    float* att   = (float*)w; w += alignUp((size_t)Etot * sizeof(float));
    (void)ws_size; (void)n_in; (void)out_size;

    const int numTiles   = N / 16;                       // 6250
    const int gemmBlk    = 128;                          // 4 waves
    const int gemmGrid   = (numTiles + 3) / 4;
    const int nodeGrid   = (N + 255) / 256;
    const int edgeGrid   = (Etot + 255) / 256;
    const int scatGrid   = (Etot + 7) / 8;               // 8 waves / 256-thread block
    const int finGrid    = (N * FOUT + 255) / 256;

    // ---------------- layer 1 ----------------
    gat_gemm_wmma_f32<<<gemmGrid, gemmBlk, 0, stream>>>(x, W1, nullptr, P, N, Fin);
    gat_node_prep<<<nodeGrid, 256, 0, stream>>>(P, as1, ad1, asrc, adst, mbuf, dens, acc, N);
    gat_edge_att<<<edgeGrid, 256, 0, stream>>>(ei, E, Etot, asrc, adst, att, mbuf);
    gat_edge_exp<<<edgeGrid, 256, 0, stream>>>(ei, E, Etot, mbuf, att, dens);
    gat_edge_scatter<<<scatGrid, 256, 0, stream>>>(ei, E, Etot, att, dens, P, acc);
    gat_finish<<<finGrid, 256, 0, stream>>>(acc, b1, Hbuf, N * FOUT);

    // ---------------- layer 2 ----------------
    gat_gemm_wmma_f32<<<gemmGrid, gemmBlk, 0, stream>>>(Hbuf, W2, nullptr, P, N, FOUT);
    gat_node_prep<<<nodeGrid, 256, 0, stream>>>(P, as2, ad2, asrc, adst, mbuf, dens, acc, N);
    gat_edge_att<<<edgeGrid, 256, 0, stream>>>(ei, E, Etot, asrc, adst, att, mbuf);
    gat_edge_exp<<<edgeGrid, 256, 0, stream>>>(ei, E, Etot, mbuf, att, dens);
    gat_edge_scatter<<<scatGrid, 256, 0, stream>>>(ei, E, Etot, att, dens, P, acc);
    gat_finish<<<finGrid, 256, 0, stream>>>(acc, b2, Hbuf, N * FOUT);

    // ---------------- readout: out = Hbuf @ Wr + br ----------------
    gat_gemm_wmma_f32<<<gemmGrid, gemmBlk, 0, stream>>>(Hbuf, Wr, br, out, N, FOUT);
}